// LPNET_V04_20109036880225
// MI455X (gfx1250) — compile-verified
//
#include <hip/hip_runtime.h>
#include <hip/hip_bf16.h>
#include <math.h>

typedef __attribute__((ext_vector_type(2))) float v2f;
typedef __attribute__((ext_vector_type(8))) float v8f;

// ---------------- CNN front-end ----------------

// conv1: in [2,179,179], w [2,2,5,5], stride 2, relu -> out [2,88,88]
__global__ void k_conv1(const float* __restrict__ in, const float* __restrict__ w,
                        const float* __restrict__ b, float* __restrict__ out) {
    int t = blockIdx.x * blockDim.x + threadIdx.x;
    if (t >= 2 * 88 * 88) return;
    int x = t % 88, y = (t / 88) % 88, oc = t / (88 * 88);
    float acc = b[oc];
    #pragma unroll
    for (int ic = 0; ic < 2; ++ic)
        #pragma unroll
        for (int ky = 0; ky < 5; ++ky)
            #pragma unroll
            for (int kx = 0; kx < 5; ++kx)
                acc += in[ic * 179 * 179 + (2 * y + ky) * 179 + (2 * x + kx)] *
                       w[oc * 50 + ic * 25 + ky * 5 + kx];
    out[t] = fmaxf(acc, 0.0f);
}

// 3x3 maxpool stride 1 VALID: [C,H,W] -> [C,H-2,W-2]
__global__ void k_pool3(const float* __restrict__ in, float* __restrict__ out,
                        int C, int H, int W) {
    int OH = H - 2, OW = W - 2;
    int t = blockIdx.x * blockDim.x + threadIdx.x;
    if (t >= C * OH * OW) return;
    int x = t % OW, y = (t / OW) % OH, c = t / (OW * OH);
    float m = -3.402823466e38f;
    #pragma unroll
    for (int ky = 0; ky < 3; ++ky)
        #pragma unroll
        for (int kx = 0; kx < 3; ++kx)
            m = fmaxf(m, in[c * H * W + (y + ky) * W + (x + kx)]);
    out[t] = m;
}

// conv2: in [2,86,86], w [1,2,3,3], stride 1, relu -> out [84,84]
__global__ void k_conv2(const float* __restrict__ in, const float* __restrict__ w,
                        const float* __restrict__ b, float* __restrict__ out) {
    int t = blockIdx.x * blockDim.x + threadIdx.x;
    if (t >= 84 * 84) return;
    int x = t % 84, y = t / 84;
    float acc = b[0];
    #pragma unroll
    for (int ic = 0; ic < 2; ++ic)
        #pragma unroll
        for (int ky = 0; ky < 3; ++ky)
            #pragma unroll
            for (int kx = 0; kx < 3; ++kx)
                acc += in[ic * 86 * 86 + (y + ky) * 86 + (x + kx)] *
                       w[ic * 9 + ky * 3 + kx];
    out[t] = fmaxf(acc, 0.0f);
}

// pred[r] = relu(f0_w[r,:] @ flat + f0_b[r]), r in [0,128), flat[6724]
__global__ void k_f0(const float* __restrict__ w, const float* __restrict__ b,
                     const float* __restrict__ x, float* __restrict__ pred) {
    __shared__ float red[256];
    int r = blockIdx.x;
    float s = 0.0f;
    for (int c = threadIdx.x; c < 6724; c += 256)
        s += w[(size_t)r * 6724 + c] * x[c];
    red[threadIdx.x] = s;
    __syncthreads();
    for (int off = 128; off > 0; off >>= 1) {
        if (threadIdx.x < (unsigned)off) red[threadIdx.x] += red[threadIdx.x + off];
        __syncthreads();
    }
    if (threadIdx.x == 0) pred[r] = fmaxf(red[0] + b[r], 0.0f);
}

// ---------------- WMMA: M = mlp2_w[2x512] @ mlp1_w[512x1152] ----------------
// One wave per 16-column output tile; A padded to 16 rows (rows 2..15 zero).
// f32 WMMA 16x16x4: A 16x4 (2 VGPR/lane), B 4x16 (2 VGPR/lane), C 16x16 (8 VGPR/lane).
// Zero-padding done branch-free: lanes with row>=2 alias row 0/1 (valid memory)
// and scale by 0, so the A fetch is an unconditional global_load_b64.
__global__ void k_matM(const float* __restrict__ m2w, const float* __restrict__ m1w,
                       float* __restrict__ M) {
    int lane = threadIdx.x;            // 0..31, one wave per block
    int tile = blockIdx.x;             // 0..71
    int n = tile * 16 + (lane & 15);   // output column
    int row = lane & 15;               // A-matrix row for this lane
    const float sel = (row < 2) ? 1.0f : 0.0f;
    const float* arow = m2w + (size_t)(row & 1) * 512;  // rows>=2 alias, zeroed by sel
    const int koff = (lane < 16) ? 0 : 2;               // K pair held by this lane
    v8f c = {};
    for (int k = 0; k < 512; k += 4) {
        int ka = k + koff;                     // even -> 8B-aligned float2
        float2 av = *(const float2*)(arow + ka);
        v2f a, bb;
        a.x = av.x * sel;
        a.y = av.y * sel;
        bb.x = m1w[(size_t)(ka)     * 1152 + n];
        bb.y = m1w[(size_t)(ka + 1) * 1152 + n];
        c = __builtin_amdgcn_wmma_f32_16x16x4_f32(
                false, a, false, bb, (short)0, c, false, false);
    }
    // C layout: lanes 0-15 hold N=lane, M = VGPR index 0..7 -> rows 0,1 are c[0],c[1]
    if (lane < 16) {
        M[0 * 1152 + tile * 16 + lane] = c[0];
        M[1 * 1152 + tile * 16 + lane] = c[1];
    }
}

// a0[r] = mlp2_b[r] + mlp2_w[r,:]@mlp1_b + M[r,:128]@pred   (r = 0,1)
__global__ void k_a0(const float* __restrict__ m2w, const float* __restrict__ m2b,
                     const float* __restrict__ m1b, const float* __restrict__ M,
                     const float* __restrict__ pred, float* __restrict__ a0) {
    int r = threadIdx.x;
    if (r < 2) {
        float s = m2b[r];
        for (int j = 0; j < 512; ++j) s += m2w[r * 512 + j] * m1b[j];
        for (int p = 0; p < 128; ++p) s += M[r * 1152 + p] * pred[p];
        a0[r] = s;
    }
}

// ---------------- Persistent sequential scan (256 steps) ----------------
// Thread j owns gi rows j, 1024+j, 2048+j (in registers). h0==0 => gh = bhh const.
// acc = a0 + M[:,128:]@h ; gi += wih[:,2i]*x + wih[:,2i+1]*y (incremental d build).
__global__ void __launch_bounds__(1024, 1)
k_scan(const float* __restrict__ wih, const float* __restrict__ bih,
       const float* __restrict__ bhh, const float* __restrict__ M,
       const float* __restrict__ a0, float* __restrict__ dout) {
    const int j = threadIdx.x;           // 0..1023
    const int lane = j & 31, wave = j >> 5;
    __shared__ float s_p0[32], s_p1[32], s_bc[2];

    float gi_r = bih[j];
    float gi_z = bih[1024 + j];
    float gi_n = bih[2048 + j];
    const float bh_r = bhh[j], bh_z = bhh[1024 + j], bh_n = bhh[2048 + j];
    const float mh0 = M[128 + j], mh1 = M[1152 + 128 + j];
    const float a00 = a0[0], a01 = a0[1];

    const float* wr0 = wih + (size_t)j * 512;
    const float* wr1 = wih + (size_t)(1024 + j) * 512;
    const float* wr2 = wih + (size_t)(2048 + j) * 512;

    float prevx = 0.f, prevy = 0.f, p2x = 0.f, p2y = 0.f;  // thread 0 state

    for (int i = 0; i < 256; ++i) {
        // prefetch upcoming wih column pairs (emits global_prefetch_b8)
        int pfc = 2 * i + 16;
        if (pfc < 512) {
            __builtin_prefetch(wr0 + pfc, 0, 1);
            __builtin_prefetch(wr1 + pfc, 0, 1);
            __builtin_prefetch(wr2 + pfc, 0, 1);
        }

        // GRU step with h_prev = 0
        float r = 1.0f / (1.0f + __expf(-(gi_r + bh_r)));
        float z = 1.0f / (1.0f + __expf(-(gi_z + bh_z)));
        float n = tanhf(gi_n + r * bh_n);
        float h = (1.0f - z) * n;

        // acc = a0 + M[:,128:] @ h  (block-wide reduction, wave32 shuffles)
        float p0 = mh0 * h, p1 = mh1 * h;
        #pragma unroll
        for (int off = 16; off > 0; off >>= 1) {
            p0 += __shfl_down(p0, off, 32);
            p1 += __shfl_down(p1, off, 32);
        }
        if (lane == 0) { s_p0[wave] = p0; s_p1[wave] = p1; }
        __syncthreads();

        if (j == 0) {
            float acc0 = a00, acc1 = a01;
            for (int wv = 0; wv < 32; ++wv) { acc0 += s_p0[wv]; acc1 += s_p1[wv]; }
            float nx = 2.0f * prevx - p2x + acc0;
            float ny = 2.0f * prevy - p2y + acc1;
            p2x = prevx; p2y = prevy; prevx = nx; prevy = ny;
            dout[2 * i] = nx;
            dout[2 * i + 1] = ny;
            s_bc[0] = nx; s_bc[1] = ny;
        }
        __syncthreads();

        // incremental gi update with the two new d elements
        float nx = s_bc[0], ny = s_bc[1];
        const int col = 2 * i;
        float2 w0 = *(const float2*)(wr0 + col);
        float2 w1 = *(const float2*)(wr1 + col);
        float2 w2 = *(const float2*)(wr2 + col);
        gi_r += w0.x * nx + w0.y * ny;
        gi_z += w1.x * nx + w1.y * ny;
        gi_n += w2.x * nx + w2.y * ny;
    }
}

// ---------------- launch ----------------

extern "C" void kernel_launch(void* const* d_in, const int* in_sizes, int n_in,
                              void* d_out, int out_size, void* d_ws, size_t ws_size,
                              hipStream_t stream) {
    (void)in_sizes; (void)n_in; (void)out_size; (void)ws_size;

    const float* emb = (const float*)d_in[0];
    const float* c1w = (const float*)d_in[1];
    const float* c1b = (const float*)d_in[2];
    const float* c2w = (const float*)d_in[3];
    const float* c2b = (const float*)d_in[4];
    const float* f0w = (const float*)d_in[5];
    const float* f0b = (const float*)d_in[6];
    const float* wih = (const float*)d_in[7];
    /* gru_whh (d_in[8]) unused: h0 == 0 every step */
    const float* bih = (const float*)d_in[9];
    const float* bhh = (const float*)d_in[10];
    const float* m1w = (const float*)d_in[11];
    const float* m1b = (const float*)d_in[12];
    const float* m2w = (const float*)d_in[13];
    const float* m2b = (const float*)d_in[14];

    float* ws = (float*)d_ws;
    float* t1 = ws;              // conv1 out   [2,88,88]  = 15488
    float* p1 = t1 + 15488;      // pool1 out   [2,86,86]  = 14792
    float* t2 = p1 + 14792;      // conv2 out   [84,84]    =  7056
    float* fl = t2 + 7056;       // pool2 out   [6724]
    float* pr = fl + 6724;       // pred        [128]
    float* M  = pr + 128;        // mlp2_w@mlp1_w [2,1152] =  2304
    float* a0 = M + 2304;        // [2]

    k_conv1<<<(2 * 88 * 88 + 255) / 256, 256, 0, stream>>>(emb, c1w, c1b, t1);
    k_pool3<<<(2 * 86 * 86 + 255) / 256, 256, 0, stream>>>(t1, p1, 2, 88, 88);
    k_conv2<<<(84 * 84 + 255) / 256, 256, 0, stream>>>(p1, c2w, c2b, t2);
    k_pool3<<<(82 * 82 + 255) / 256, 256, 0, stream>>>(t2, fl, 1, 84, 84);
    k_f0   <<<128, 256, 0, stream>>>(f0w, f0b, fl, pr);
    k_matM <<<72, 32, 0, stream>>>(m2w, m1w, M);
    k_a0   <<<1, 32, 0, stream>>>(m2w, m2b, m1b, M, pr, a0);
    k_scan <<<1, 1024, 0, stream>>>(wih, bih, bhh, M, a0, (float*)d_out);
}